// ChamferLoss_32581621908071
// MI455X (gfx1250) — compile-verified
//
#include <hip/hip_runtime.h>
#include <hip/hip_bf16.h>

typedef __attribute__((ext_vector_type(2))) float v2f;
typedef __attribute__((ext_vector_type(8))) float v8f;

#define BATCH 32
#define NPTS 2048   // x points per batch
#define MPTS 2048   // y points per batch
#define WAVES 8
#define TILES_I (MPTS / 16)  // 128
#define TILES_J (NPTS / 16)  // 128

__global__ __launch_bounds__(256) void chamfer_main_kernel(
    const float* __restrict__ x,   // [B, N, 3]
    const float* __restrict__ y,   // [B, M, 3]
    float* __restrict__ ws)        // [B] per-batch partial results
{
  __shared__ int    xmin_lds[NPTS];            // x_min per x-point (float bits, nonneg)
  __shared__ float2 bfrag_lds[TILES_J * 32];   // per-(jtile,lane) B fragment
  __shared__ float  nx_lds[NPTS];              // |x|^2 per x-point
  __shared__ float  ysum_lds[WAVES];
  __shared__ float  xsum_lds[WAVES];

  const int b    = blockIdx.x;
  const int tid  = threadIdx.x;
  const int wave = tid >> 5;
  const int lane = tid & 31;
  const int half = lane >> 4;   // 0: lanes 0-15, 1: lanes 16-31
  const int lm   = lane & 15;

  const float* xb = x + (size_t)b * NPTS * 3;
  const float* yb = y + (size_t)b * MPTS * 3;

  // ---- init x_min to +inf bits ----
  for (int i = tid; i < NPTS; i += 256) xmin_lds[i] = 0x7F800000;

  // ---- stage X: precompute B fragments and |x|^2 into LDS (once per block) ----
  for (int jt = wave; jt < TILES_J; jt += WAVES) {
    const int j0 = jt * 16;
    const float x0 = xb[(j0 + lm) * 3 + 0];
    const float x1 = xb[(j0 + lm) * 3 + 1];
    const float x2 = xb[(j0 + lm) * 3 + 2];
    float2 bf;
    bf.x = half ? x2 : x0;    // K=2 : K=0
    bf.y = half ? 0.0f : x1;  // K=3(pad) : K=1
    bfrag_lds[jt * 32 + lane] = bf;
    if (half == 0) nx_lds[j0 + lm] = x0 * x0 + x1 * x1 + x2 * x2;
  }
  __syncthreads();

  float ysum = 0.0f;  // meaningful on lanes with lm==0 (lanes 0 and 16)

  for (int it = wave; it < TILES_I; it += WAVES) {
    const int i0 = it * 16;

    // Load Y point for row i0+lm; build A fragment (16x4 f32, K=3 padded to 4)
    const float y0 = yb[(i0 + lm) * 3 + 0];
    const float y1 = yb[(i0 + lm) * 3 + 1];
    const float y2 = yb[(i0 + lm) * 3 + 2];
    v2f afrag;
    afrag.x = half ? y2 : y0;
    afrag.y = half ? 0.0f : y1;
    const float ny = y0 * y0 + y1 * y1 + y2 * y2;

    // ny for row (v + 8*half) lives in lane (v + 8*half)
    float rowny[8];
#pragma unroll
    for (int v = 0; v < 8; ++v)
      rowny[v] = __shfl(ny, v + 8 * half, 32);

    float ymin[8];
#pragma unroll
    for (int v = 0; v < 8; ++v) ymin[v] = __builtin_inff();

#pragma unroll 2
    for (int jt = 0; jt < TILES_J; ++jt) {
      const int j0 = jt * 16;
      const float2 bf = bfrag_lds[jt * 32 + lane];
      const float  nx = nx_lds[j0 + lm];
      v2f bfrag;
      bfrag.x = bf.x;
      bfrag.y = bf.y;

      // G tile = Y_tile (16x4) * X_tile^T (4x16), f32 WMMA
      v8f c = {};
      c = __builtin_amdgcn_wmma_f32_16x16x4_f32(
          /*neg_a=*/false, afrag, /*neg_b=*/false, bfrag,
          /*c_mod=*/(short)0, c, /*reuse_a=*/false, /*reuse_b=*/false);

      float xmin_l = __builtin_inff();
#pragma unroll
      for (int v = 0; v < 8; ++v) {
        // element (m = v + 8*half, n = lm): d = |y_m|^2 + |x_n|^2 - 2*G
        float dval = rowny[v] + nx - 2.0f * c[v];
        dval = fmaxf(dval, 0.0f);  // true distance >= 0; enables int-bit min
        ymin[v]  = fminf(ymin[v], dval);
        xmin_l   = fminf(xmin_l, dval);
      }
      // lanes l and l+16 hit the same address; min is commutative, so let
      // both halves issue the atomic (no shuffle, no EXEC divergence)
      atomicMin(&xmin_lds[j0 + lm], __float_as_int(xmin_l));
    }

    // y_min: reduce over the 16 lanes (columns) within each half-wave
#pragma unroll
    for (int v = 0; v < 8; ++v) {
      float m = ymin[v];
      m = fminf(m, __shfl_xor(m, 8, 32));
      m = fminf(m, __shfl_xor(m, 4, 32));
      m = fminf(m, __shfl_xor(m, 2, 32));
      m = fminf(m, __shfl_xor(m, 1, 32));
      ymin[v] = m;
    }
    if (lm == 0) {
      float s = 0.0f;
#pragma unroll
      for (int v = 0; v < 8; ++v) s += ymin[v];
      ysum += s;  // lane 0: rows v; lane 16: rows v+8
    }
  }

  // combine lane 0 + lane 16 partial y sums, stash per wave
  float ys = ysum + __shfl_xor(ysum, 16, 32);
  if (lane == 0) ysum_lds[wave] = ys;
  __syncthreads();

  // sum x_min over all N points
  float xs = 0.0f;
  for (int i = tid; i < NPTS; i += 256) xs += __int_as_float(xmin_lds[i]);
  for (int off = 16; off >= 1; off >>= 1) xs += __shfl_xor(xs, off, 32);
  if (lane == 0) xsum_lds[wave] = xs;
  __syncthreads();

  if (tid == 0) {
    float xt = 0.0f, yt = 0.0f;
    for (int w = 0; w < WAVES; ++w) { xt += xsum_lds[w]; yt += ysum_lds[w]; }
    ws[b] = xt / (float)NPTS + yt / (float)MPTS;
  }
}

__global__ void chamfer_final_kernel(const float* __restrict__ ws,
                                     float* __restrict__ out) {
  // single wave of 32, one value per batch
  float v = ws[threadIdx.x];
  for (int off = 16; off >= 1; off >>= 1) v += __shfl_xor(v, off, 32);
  if (threadIdx.x == 0) out[0] = v / (float)BATCH;
}

extern "C" void kernel_launch(void* const* d_in, const int* in_sizes, int n_in,
                              void* d_out, int out_size, void* d_ws, size_t ws_size,
                              hipStream_t stream) {
  const float* x = (const float*)d_in[0];  // [32, 2048, 3]
  const float* y = (const float*)d_in[1];  // [32, 2048, 3]
  float* out = (float*)d_out;              // scalar
  float* ws  = (float*)d_ws;               // 32 floats of scratch

  chamfer_main_kernel<<<BATCH, 256, 0, stream>>>(x, y, ws);
  chamfer_final_kernel<<<1, 32, 0, stream>>>(ws, out);
}